// TtMambaBlock_67843303407623
// MI455X (gfx1250) — compile-verified
//
#include <hip/hip_runtime.h>
#include <hip/hip_bf16.h>
#include <stdint.h>

#define USERS   32
#define DMODEL  2560
#define DINNER  5120
#define DSTATE  16
#define DTRANK  160
#define XDBL_N  (DTRANK + 2 * DSTATE)   // 192
#define OUT_SPLIT 4
#define XP_SPLIT  32

typedef __attribute__((ext_vector_type(2))) float v2f;
typedef __attribute__((ext_vector_type(8))) float v8f;
typedef unsigned int u32x4 __attribute__((ext_vector_type(4)));
typedef int          i32x4 __attribute__((ext_vector_type(4)));
typedef int          i32x8 __attribute__((ext_vector_type(8)));

// ---------------------------------------------------------------------------
// Tensor Data Mover: queue a 2D tile DMA (rows x cols fp32, row stride
// `stride_elems`) from global into LDS at byte offset lds_off.
// D# layout per CDNA5 ISA 8.3/8.4: group0 = {count|flags, lds_addr,
// global_addr[31:0], global_addr[56:32]|type=2}; group1 carries data_size=4B,
// tensor dims, tile dims, dim0 stride. Groups 2/3 zero (2D tile: tile_dim2=0
// terminates the walker). Tracked with TENSORcnt.
// ---------------------------------------------------------------------------
__device__ __forceinline__ void tdm_load_2d(const float* gptr, uint32_t lds_off,
                                            int rows, int cols, int stride_elems)
{
    const uint64_t ga = (uint64_t)(uintptr_t)gptr;
    const uint32_t sd = (uint32_t)stride_elems;
    const uint32_t td1 = 1u << 24;                 // tensor_dim1 (never OOB)
    u32x4 g0;
    g0[0] = 1u;                                    // count=1, user descriptor
    g0[1] = lds_off;                               // lds_addr (bytes)
    g0[2] = (uint32_t)ga;                          // global_addr[31:0]
    g0[3] = (uint32_t)((ga >> 32) & 0x01ffffffu)   // global_addr[56:32]
          | (2u << 30);                            // type = 2 ("image")
    i32x8 g1;
    g1[0] = (int)(2u << 16);                       // data_size = 2 -> 4 bytes
    g1[1] = (int)((sd & 0xffffu) << 16);           // tensor_dim0[15:0]=stride
    g1[2] = (int)(((sd >> 16) & 0xffffu)           // tensor_dim0[31:16]
          | ((td1 & 0xffffu) << 16));              // tensor_dim1[15:0]
    g1[3] = (int)(((td1 >> 16) & 0xffffu)          // tensor_dim1[31:16]
          | (((uint32_t)cols & 0xffffu) << 16));   // tile_dim0
    g1[4] = (int)((uint32_t)rows & 0xffffu);       // tile_dim1 (tile_dim2=0)
    g1[5] = (int)sd;                               // tensor_dim0_stride[31:0]
    g1[6] = 0;                                     // stride[47:32], dim1_stride lo
    g1[7] = 0;
    i32x4 z = {0, 0, 0, 0};
#if __clang_major__ >= 23
    i32x8 z8 = {0, 0, 0, 0, 0, 0, 0, 0};
    __builtin_amdgcn_tensor_load_to_lds(g0, g1, z, z, z8, 0);
#else
    __builtin_amdgcn_tensor_load_to_lds(g0, g1, z, z, 0);
#endif
}

// ---------------------------------------------------------------------------
// One wave computes a 32x16 strip of C = A(32xK) * W(KxN) with
// V_WMMA_F32_16X16X4_F32. Weights are streamed by the TDM into a per-wave
// double-buffered LDS tile (KC x 16 fp32); compute reads B fragments from LDS
// (ds_load) while the next tile is in flight (TENSORcnt double buffering).
// A (32xK activations, L2-resident) is read directly from global.
//
// Fragment layout (wave32, ISA 7.12.2):
//   A 16x4 f32:  lane l -> row (l&15), K = 2*(l>>4)+{0,1}
//   B 4x16 f32:  lane l -> col (l&15), K = 2*(l>>4)+{0,1}
//   C 16x16 f32: lane l, vgpr r -> row r+8*(l>>4), col (l&15)
// ---------------------------------------------------------------------------
template <int KC>
__device__ __forceinline__ void gemm32_tdm(const float* __restrict__ A, int lda,
                                           const float* __restrict__ W, int ldw,
                                           int k0, int kLen, int n0,
                                           v8f& accLo, v8f& accHi)
{
    __shared__ float smem[4][2][KC * 16];          // 4 waves / block
    const int wv   = __builtin_amdgcn_readfirstlane((int)(threadIdx.x >> 5));
    const int lane = threadIdx.x & 31;
    const int sub  = lane & 15;
    const int koff = (lane >> 4) << 1;             // 0 or 2
    const float* aLo = A + (size_t)sub * lda + koff;
    const float* aHi = A + (size_t)(sub + 16) * lda + koff;

    const uint32_t lds0 = (uint32_t)(uintptr_t)&smem[wv][0][0];
    const float* wtile  = W + (size_t)k0 * ldw + n0;
    const int nCh = kLen / KC;

    tdm_load_2d(wtile, lds0, KC, 16, ldw);         // prologue: chunk 0
    for (int c = 0; c < nCh; ++c) {
        if (c + 1 < nCh) {
            // LDS reads of the buffer we are about to overwrite are long
            // retired (consumed by WMMA), but order it architecturally.
            asm volatile("s_wait_dscnt 0x0" ::: "memory");
            tdm_load_2d(wtile + (size_t)(c + 1) * KC * ldw,
                        lds0 + (uint32_t)(((c + 1) & 1) * (KC * 16 * 4)),
                        KC, 16, ldw);
            __builtin_amdgcn_s_wait_tensorcnt((short)1);   // chunk c landed
        } else {
            __builtin_amdgcn_s_wait_tensorcnt((short)0);
        }
        asm volatile("" ::: "memory");             // fence LDS reads vs TDM
        const float* wt = &smem[wv][c & 1][0];
        const int kc = k0 + c * KC;
#pragma unroll
        for (int kk = 0; kk < KC; kk += 4) {
            v2f a0 = *(const v2f*)(aLo + kc + kk);
            v2f a1 = *(const v2f*)(aHi + kc + kk);
            v2f b;
            b.x = wt[(kk + koff)     * 16 + sub];  // ds_load from staged tile
            b.y = wt[(kk + koff + 1) * 16 + sub];
            accLo = __builtin_amdgcn_wmma_f32_16x16x4_f32(
                        false, a0, false, b, (short)0, accLo, false, false);
            accHi = __builtin_amdgcn_wmma_f32_16x16x4_f32(
                        false, a1, false, b, (short)0, accHi, false, false);
        }
    }
}

// ---------------------------------------------------------------------------
// Kernel 1: xs = xb @ w_in_ssm (fused conv4 + SiLU -> cbuf)
//           res = xb @ w_in_mlp (-> resbuf).  640 waves.
// ---------------------------------------------------------------------------
__global__ void __launch_bounds__(128)
k_inproj_conv(const float* __restrict__ xb,
              const float* __restrict__ w_ssm,
              const float* __restrict__ w_mlp,
              const float* __restrict__ conv_w,
              const float* __restrict__ conv_b,
              const float* __restrict__ cs1,
              const float* __restrict__ cs2,
              const float* __restrict__ cs3,
              float* __restrict__ cbuf,
              float* __restrict__ resbuf)
{
    const int wave = __builtin_amdgcn_readfirstlane(
        (int)((blockIdx.x * blockDim.x + threadIdx.x) >> 5));
    const int sel  = wave / 320;              // 0 = ssm branch, 1 = mlp branch
    const int n0   = (wave % 320) * 16;
    const float* W = sel ? w_mlp : w_ssm;

    v8f accLo = {}; v8f accHi = {};
    gemm32_tdm<64>(xb, DMODEL, W, DINNER, 0, DMODEL, n0, accLo, accHi);

    const int lane = threadIdx.x & 31;
    const int col  = n0 + (lane & 15);
    const int rb   = (lane >> 4) << 3;
    if (sel) {
#pragma unroll
        for (int r = 0; r < 8; ++r) {
            resbuf[(size_t)(rb + r)      * DINNER + col] = accLo[r];
            resbuf[(size_t)(rb + r + 16) * DINNER + col] = accHi[r];
        }
    } else {
        const float w0 = conv_w[col];
        const float w1 = conv_w[DINNER + col];
        const float w2 = conv_w[2 * DINNER + col];
        const float w3 = conv_w[3 * DINNER + col];
        const float cb = conv_b[col];
#pragma unroll
        for (int r = 0; r < 8; ++r) {
            const int u0 = rb + r, u1 = rb + r + 16;
            float v0 = cs1[(size_t)u0 * DINNER + col] * w0
                     + cs2[(size_t)u0 * DINNER + col] * w1
                     + cs3[(size_t)u0 * DINNER + col] * w2
                     + accLo[r] * w3 + cb;
            float v1 = cs1[(size_t)u1 * DINNER + col] * w0
                     + cs2[(size_t)u1 * DINNER + col] * w1
                     + cs3[(size_t)u1 * DINNER + col] * w2
                     + accHi[r] * w3 + cb;
            cbuf[(size_t)u0 * DINNER + col] = v0 / (1.f + __expf(-v0)); // SiLU
            cbuf[(size_t)u1 * DINNER + col] = v1 / (1.f + __expf(-v1));
        }
    }
}

// ---------------------------------------------------------------------------
// Kernel 2a: x_dbl partials = c @ x_proj_w, split-K x32 -> 32 partial planes
// (deterministic; no float atomics). 384 waves.
// ---------------------------------------------------------------------------
__global__ void __launch_bounds__(128)
k_xproj(const float* __restrict__ cbuf,
        const float* __restrict__ xpw,
        float* __restrict__ xpart)          // [XP_SPLIT][USERS*XDBL_N]
{
    const int wave = __builtin_amdgcn_readfirstlane(
        (int)((blockIdx.x * blockDim.x + threadIdx.x) >> 5));
    const int n0    = (wave % 12) * 16;
    const int chunk = wave / 12;
    const int kChnk = DINNER / XP_SPLIT;           // 160

    v8f accLo = {}; v8f accHi = {};
    gemm32_tdm<32>(cbuf, DINNER, xpw, XDBL_N, chunk * kChnk, kChnk, n0,
                   accLo, accHi);

    float* plane = xpart + (size_t)chunk * (USERS * XDBL_N);
    const int lane = threadIdx.x & 31;
    const int col  = n0 + (lane & 15);
    const int rb   = (lane >> 4) << 3;
#pragma unroll
    for (int r = 0; r < 8; ++r) {
        plane[(size_t)(rb + r)      * XDBL_N + col] = accLo[r];
        plane[(size_t)(rb + r + 16) * XDBL_N + col] = accHi[r];
    }
}

// Kernel 2b: reduce the 32 partial planes into x_dbl (32x192).
__global__ void __launch_bounds__(256)
k_xred(const float* __restrict__ xpart, float* __restrict__ x_dbl)
{
    const int i = (int)(blockIdx.x * blockDim.x + threadIdx.x);
    if (i >= USERS * XDBL_N) return;
    float s = 0.f;
#pragma unroll
    for (int p = 0; p < XP_SPLIT; ++p)
        s += xpart[(size_t)p * (USERS * XDBL_N) + i];
    x_dbl[i] = s;
}

// ---------------------------------------------------------------------------
// Kernel 3: dt = softplus(x_dbl[:, :160] @ dt_proj_w + dt_proj_b). 320 waves.
// ---------------------------------------------------------------------------
__global__ void __launch_bounds__(128)
k_dtproj(const float* __restrict__ x_dbl,
         const float* __restrict__ dtw,
         const float* __restrict__ dtb,
         float* __restrict__ dt)
{
    const int wave = __builtin_amdgcn_readfirstlane(
        (int)((blockIdx.x * blockDim.x + threadIdx.x) >> 5));
    const int n0 = wave * 16;

    v8f accLo = {}; v8f accHi = {};
    gemm32_tdm<32>(x_dbl, XDBL_N, dtw, DINNER, 0, DTRANK, n0, accLo, accHi);

    const int lane = threadIdx.x & 31;
    const int col  = n0 + (lane & 15);
    const int rb   = (lane >> 4) << 3;
    const float b  = dtb[col];
#pragma unroll
    for (int r = 0; r < 8; ++r) {
        float v0 = accLo[r] + b;
        float v1 = accHi[r] + b;
        v0 = (v0 > 20.f) ? v0 : __logf(1.f + __expf(v0));   // softplus
        v1 = (v1 > 20.f) ? v1 : __logf(1.f + __expf(v1));
        dt[(size_t)(rb + r)      * DINNER + col] = v0;
        dt[(size_t)(rb + r + 16) * DINNER + col] = v1;
    }
}

// ---------------------------------------------------------------------------
// Kernel 4: selective-scan state update + gate. One thread per (user,channel);
// 16 contiguous f32 states -> 64B-aligned b128 loads. zbuf = y * res.
// ---------------------------------------------------------------------------
__global__ void __launch_bounds__(256)
k_ssm(const float* __restrict__ dtbuf,
      const float* __restrict__ cbuf,
      const float* __restrict__ resbuf,
      const float* __restrict__ x_dbl,
      const float* __restrict__ A_log,
      const float* __restrict__ D_param,
      const float* __restrict__ ssm_state,
      float* __restrict__ zbuf)
{
    const int idx = (int)(blockIdx.x * blockDim.x + threadIdx.x);
    if (idx >= USERS * DINNER) return;
    const int u = idx / DINNER;
    const int d = idx - u * DINNER;

    const float dtv = dtbuf[idx];
    const float cv  = cbuf[idx];
    const float rv  = resbuf[idx];
    const float dtc = dtv * cv;

    const float* st = ssm_state + (size_t)idx * DSTATE;
    const float* Ar = A_log     + (size_t)d   * DSTATE;
    const float* Bp = x_dbl + (size_t)u * XDBL_N + DTRANK;
    const float* Cp = Bp + DSTATE;

    float y = 0.f;
#pragma unroll
    for (int s = 0; s < DSTATE; ++s) {
        const float a  = -__expf(Ar[s]);
        const float dA = __expf(dtv * a);
        const float ns = st[s] * dA + dtc * Bp[s];
        y += ns * Cp[s];
    }
    y += D_param[d] * cv;
    zbuf[idx] = y * rv;
}

// ---------------------------------------------------------------------------
// Kernel 5a: out partials = z @ w_out, split-K x4 (640 waves streaming the
// 52 MB weight). Kernel 5b reduces 4 partial planes into d_out.
// ---------------------------------------------------------------------------
__global__ void __launch_bounds__(128)
k_outproj(const float* __restrict__ zbuf,
          const float* __restrict__ w_out,
          float* __restrict__ opart)        // [OUT_SPLIT][USERS*DMODEL]
{
    const int wave = __builtin_amdgcn_readfirstlane(
        (int)((blockIdx.x * blockDim.x + threadIdx.x) >> 5));
    const int n0 = (wave % 160) * 16;
    const int s  = wave / 160;
    const int kC = DINNER / OUT_SPLIT;             // 1280

    v8f accLo = {}; v8f accHi = {};
    gemm32_tdm<64>(zbuf, DINNER, w_out, DMODEL, s * kC, kC, n0, accLo, accHi);

    float* plane = opart + (size_t)s * (USERS * DMODEL);
    const int lane = threadIdx.x & 31;
    const int col  = n0 + (lane & 15);
    const int rb   = (lane >> 4) << 3;
#pragma unroll
    for (int r = 0; r < 8; ++r) {
        plane[(size_t)(rb + r)      * DMODEL + col] = accLo[r];
        plane[(size_t)(rb + r + 16) * DMODEL + col] = accHi[r];
    }
}

__global__ void __launch_bounds__(256)
k_outred(const float* __restrict__ opart, float* __restrict__ out)
{
    const int i = (int)(blockIdx.x * blockDim.x + threadIdx.x);
    if (i >= USERS * DMODEL) return;
    float s = 0.f;
#pragma unroll
    for (int p = 0; p < OUT_SPLIT; ++p)
        s += opart[(size_t)p * (USERS * DMODEL) + i];
    out[i] = s;
}

// ---------------------------------------------------------------------------
extern "C" void kernel_launch(void* const* d_in, const int* in_sizes, int n_in,
                              void* d_out, int out_size, void* d_ws, size_t ws_size,
                              hipStream_t stream)
{
    const float* xb     = (const float*)d_in[0];
    const float* w_ssm  = (const float*)d_in[1];
    const float* w_mlp  = (const float*)d_in[2];
    const float* w_out  = (const float*)d_in[3];
    const float* convw  = (const float*)d_in[4];
    const float* convb  = (const float*)d_in[5];
    const float* cs1    = (const float*)d_in[6];
    const float* cs2    = (const float*)d_in[7];
    const float* cs3    = (const float*)d_in[8];
    const float* xpw    = (const float*)d_in[9];
    const float* dtw    = (const float*)d_in[10];
    const float* dtb    = (const float*)d_in[11];
    const float* Alog   = (const float*)d_in[12];
    const float* Dp     = (const float*)d_in[13];
    const float* sstate = (const float*)d_in[14];
    (void)in_sizes; (void)n_in; (void)out_size; (void)ws_size;

    const size_t PLANE = (size_t)USERS * DINNER * sizeof(float);   // 655360 B
    char* ws = (char*)d_ws;
    float* cbuf  = (float*)(ws + 0 * PLANE);
    float* resb  = (float*)(ws + 1 * PLANE);
    float* dtbuf = (float*)(ws + 2 * PLANE);
    float* zbuf  = (float*)(ws + 3 * PLANE);
    float* xdbl  = (float*)(ws + 4 * PLANE);                       // 24 KB
    float* xpart = (float*)(ws + 4 * PLANE + 32768);               // 768 KB
    float* opart = (float*)(ws + 4 * PLANE + 32768 +
                            (size_t)XP_SPLIT * USERS * XDBL_N * sizeof(float));

    k_inproj_conv<<<160, 128, 0, stream>>>(xb, w_ssm, w_mlp, convw, convb,
                                           cs1, cs2, cs3, cbuf, resb);
    k_xproj  <<< 96, 128, 0, stream>>>(cbuf, xpw, xpart);
    k_xred   <<< 24, 256, 0, stream>>>(xpart, xdbl);
    k_dtproj <<< 80, 128, 0, stream>>>(xdbl, dtw, dtb, dtbuf);
    k_ssm    <<<640, 256, 0, stream>>>(dtbuf, cbuf, resb, xdbl, Alog, Dp,
                                       sstate, zbuf);
    k_outproj<<<160, 128, 0, stream>>>(zbuf, w_out, opart);
    k_outred <<<320, 256, 0, stream>>>(opart, (float*)d_out);
}